// QuantizationLoss_69114613728896
// MI455X (gfx1250) — compile-verified
//
#include <hip/hip_runtime.h>

typedef __attribute__((ext_vector_type(2))) float v2f;
typedef __attribute__((ext_vector_type(4))) float v4f;
typedef __attribute__((ext_vector_type(8))) float v8f;

#define BLOCK 256
#define NBLOCKS 2048

// ---- quantization distance: dist = min(|x - nearest_level(x)|, 1.0) ----
__device__ __forceinline__ float qdist(float x) {
    float t = (x + 1.0f) * 127.5f;
    float r = rintf(t);                       // v_rndne_f32: round half-to-even (matches jnp.round)
    r = fminf(fmaxf(r, 0.0f), 255.0f);        // clip to [0, 255]
    float nearest = r * (1.0f / 127.5f) - 1.0f;
    return fminf(fabsf(x - nearest), 1.0f);
}

// ---- wave32 reduction via two chained V_WMMA_F32_16X16X4_F32 ----
// WMMA #1: A(16x4) lane L<16 -> A[L][0]=acc, lane L>=16 -> A[L-16][2]=acc; B = ones.
//          D1[m][n] = acc[m] + acc[m+16] for every column n.
// local8  = sum of the 8 C/D VGPRs: lanes 0-15 get S_low = sum(rows 0-7),
//           lanes 16-31 get S_high = sum(rows 8-15).
// WMMA #2: feed local8 back as A with ones B: D2[m][n] = S_low + S_high = wave total,
//          identical in every lane -> no cross-lane shuffle needed at all.
__device__ __forceinline__ float wave_reduce_wmma(float acc) {
    v2f b; b.x = 1.0f; b.y = 1.0f;            // all-ones B is layout-agnostic
    v2f a; a.x = acc;  a.y = 0.0f;
    v8f c = {};
    c = __builtin_amdgcn_wmma_f32_16x16x4_f32(false, a, false, b, (short)0, c, false, false);
    float local8 = ((c[0] + c[1]) + (c[2] + c[3])) + ((c[4] + c[5]) + (c[6] + c[7]));
    v2f a2; a2.x = local8; a2.y = 0.0f;
    v8f c2 = {};
    c2 = __builtin_amdgcn_wmma_f32_16x16x4_f32(false, a2, false, b, (short)0, c2, false, false);
    return c2[0];                              // wave total, present in all 32 lanes
}

__device__ __forceinline__ float block_reduce(float acc, float* lds) {
    float wt = wave_reduce_wmma(acc);          // EXEC all-1s here: uniform control flow
    int lane = threadIdx.x & 31;
    int wid  = threadIdx.x >> 5;
    if (lane == 0) lds[wid] = wt;
    __syncthreads();
    float s = 0.0f;
    if (threadIdx.x == 0) {
        #pragma unroll
        for (int w = 0; w < (BLOCK >> 5); ++w) s += lds[w];
    }
    return s;                                  // valid in thread 0 only
}

// ---- stage 1: stream N/4 float4s (non-temporal), one partial per block ----
__global__ __launch_bounds__(BLOCK) void qloss_stage1(const float* __restrict__ in,
                                                      float* __restrict__ partial,
                                                      long long n4) {
    const v4f* in4 = (const v4f*)in;
    float acc0 = 0.0f, acc1 = 0.0f;
    const long long stride = (long long)gridDim.x * BLOCK;
    long long i = (long long)blockIdx.x * BLOCK + threadIdx.x;

    // 2 outstanding b128 loads per iteration, independent accumulator chains
    for (; i + stride < n4; i += 2 * stride) {
        v4f x0 = __builtin_nontemporal_load(in4 + i);           // global_load_b128 ... NT
        v4f x1 = __builtin_nontemporal_load(in4 + i + stride);
        acc0 += qdist(x0[0]); acc1 += qdist(x1[0]);
        acc0 += qdist(x0[1]); acc1 += qdist(x1[1]);
        acc0 += qdist(x0[2]); acc1 += qdist(x1[2]);
        acc0 += qdist(x0[3]); acc1 += qdist(x1[3]);
    }
    if (i < n4) {
        v4f x0 = __builtin_nontemporal_load(in4 + i);
        acc0 += qdist(x0[0]); acc0 += qdist(x0[1]);
        acc0 += qdist(x0[2]); acc0 += qdist(x0[3]);
    }

    __shared__ float lds[BLOCK >> 5];
    float s = block_reduce(acc0 + acc1, lds);
    if (threadIdx.x == 0) partial[blockIdx.x] = s;
}

// ---- stage 2: single block folds the per-block partials (+ scalar tail) -> mean ----
__global__ __launch_bounds__(BLOCK) void qloss_stage2(const float* __restrict__ partial,
                                                      int nparts,
                                                      const float* __restrict__ in,
                                                      long long tail_start, long long n,
                                                      float* __restrict__ out, float inv_n) {
    float acc = 0.0f;
    for (int i = threadIdx.x; i < nparts; i += BLOCK) acc += partial[i];
    for (long long i = tail_start + threadIdx.x; i < n; i += BLOCK) acc += qdist(in[i]);
    __shared__ float lds[BLOCK >> 5];
    float s = block_reduce(acc, lds);
    if (threadIdx.x == 0) out[0] = s * inv_n;
}

extern "C" void kernel_launch(void* const* d_in, const int* in_sizes, int n_in,
                              void* d_out, int out_size, void* d_ws, size_t ws_size,
                              hipStream_t stream) {
    (void)n_in; (void)out_size;
    const float* x = (const float*)d_in[0];
    long long n  = (long long)in_sizes[0];        // 64*1*1024*1024 = 67,108,864
    long long n4 = n >> 2;
    float* partial = (float*)d_ws;

    // clamp grid to available scratch (one float partial per block)
    int nblocks = NBLOCKS;
    long long ws_cap = (long long)(ws_size / sizeof(float));
    if (ws_cap < nblocks) nblocks = (ws_cap > 0) ? (int)ws_cap : 1;

    qloss_stage1<<<nblocks, BLOCK, 0, stream>>>(x, partial, n4);
    qloss_stage2<<<1, BLOCK, 0, stream>>>(partial, nblocks, x, n4 << 2, n,
                                          (float*)d_out, 1.0f / (float)n);
}